// EdgeClassifier_5609227288774
// MI455X (gfx1250) — compile-verified
//
#include <hip/hip_runtime.h>

#define NN 50000
#define EE 400000
#define CHUNK 100
#define OUT 96
#define HID 192
#define EPS 1e-5f

typedef __attribute__((ext_vector_type(2))) float v2f;
typedef __attribute__((ext_vector_type(8))) float v8f;

__device__ __forceinline__ v8f wmma4(v2f a, v2f b, v8f c) {
  // D = A(16x4 f32) * B(4x16 f32) + C(16x16 f32)
  return __builtin_amdgcn_wmma_f32_16x16x4_f32(
      /*neg_a=*/false, a, /*neg_b=*/false, b,
      /*c_mod=*/(short)0, c, /*reuse_a=*/false, /*reuse_b=*/false);
}

// ---------------- CSR build (counting sort by dst) ----------------

__global__ void zero_i_kernel(int* __restrict__ p, int n) {
  int i = blockIdx.x * blockDim.x + threadIdx.x;
  if (i < n) p[i] = 0;
}

__global__ void degi_kernel(const int* __restrict__ dst, int* __restrict__ degi) {
  int e = blockIdx.x * blockDim.x + threadIdx.x;
  if (e < EE) atomicAdd(&degi[dst[e]], 1);
}

__global__ void norm_kernel(const int* __restrict__ degi, float* __restrict__ nrm) {
  int i = blockIdx.x * blockDim.x + threadIdx.x;
  if (i < NN) nrm[i] = degi[i] > 0 ? 1.0f / (float)degi[i] : 0.0f;
}

// single-workgroup exclusive prefix scan over degi -> rowStart[0..NN]
__global__ __launch_bounds__(1024) void scan_kernel(const int* __restrict__ degi,
                                                    int* __restrict__ rowStart) {
  __shared__ int s[1024];
  __shared__ int carry;
  const int t = threadIdx.x;
  if (t == 0) carry = 0;
  __syncthreads();
  for (int base = 0; base < NN; base += 1024) {
    const int i = base + t;
    const int x = (i < NN) ? degi[i] : 0;
    s[t] = x;
    __syncthreads();
    for (int off = 1; off < 1024; off <<= 1) {
      int val = 0;
      if (t >= off) val = s[t - off];
      __syncthreads();
      if (t >= off) s[t] += val;
      __syncthreads();
    }
    const int excl = s[t] - x;
    const int c = carry;
    if (i < NN) rowStart[i] = c + excl;
    __syncthreads();
    if (t == 0) carry += s[1023];
    __syncthreads();
  }
  if (t == 0) rowStart[NN] = carry;  // == EE
}

__global__ void fill_kernel(const int* __restrict__ src, const int* __restrict__ dst,
                            const float* __restrict__ dist,
                            const int* __restrict__ rowStart, int* __restrict__ cursor,
                            int* __restrict__ csr_src, float* __restrict__ csr_w) {
  int e = blockIdx.x * blockDim.x + threadIdx.x;
  if (e < EE) {
    int d = dst[e];
    int pos = rowStart[d] + atomicAdd(&cursor[d], 1);
    csr_src[pos] = src[e];
    csr_w[pos] = dist[e];
  }
}

// ---------------- gather: ah[v][j] = sum_k hh[csr_src[k]][j] * csr_w[k] ----------------
// grid NN, block 192 (one feature per thread, coalesced row reads)

__global__ __launch_bounds__(HID) void gather_kernel(
    const float* __restrict__ hh, const int* __restrict__ rowStart,
    const int* __restrict__ csr_src, const float* __restrict__ csr_w,
    float* __restrict__ ah) {
  const int v = blockIdx.x;
  const int j = threadIdx.x;
  const int beg = rowStart[v], end = rowStart[v + 1];
  float acc = 0.0f;
  for (int k = beg; k < end; ++k) {
    acc += hh[csr_src[k] * HID + j] * csr_w[k];
  }
  ah[v * HID + j] = acc;
}

// ---------------- input projector: per-chunk Linear(100,96)+LN+LeakyReLU ----------------
// grid (NN/16, 2), block 32 (one wave per 16-node tile per chunk)

__global__ __launch_bounds__(32) void proj_kernel(
    const float* __restrict__ h, const float* __restrict__ W,
    const float* __restrict__ bias, const float* __restrict__ g,
    const float* __restrict__ beta, float* __restrict__ hh) {
  __shared__ alignas(16) float sA[16 * CHUNK];  // 16 x 100 input tile
  __shared__ float sO[16 * OUT];                // 16 x 96 output tile
  const int lane = threadIdx.x;
  const int lo = lane & 15;
  const int hi = lane >> 4;
  const int nb = blockIdx.x * 16;  // node base
  const int c = blockIdx.y;        // chunk

  // stage 16x100 tile with b128 loads (rows are 16B-aligned: 200*4 and 100*4)
  for (int idx = lane; idx < 16 * (CHUNK / 4); idx += 32) {
    int r = idx / (CHUNK / 4), q = idx % (CHUNK / 4);
    const float4 val =
        *reinterpret_cast<const float4*>(&h[(nb + r) * (2 * CHUNK) + c * CHUNK + q * 4]);
    *reinterpret_cast<float4*>(&sA[r * CHUNK + q * 4]) = val;
  }
  __syncthreads();

  const float* Wc = W + c * CHUNK * OUT;
  for (int nt = 0; nt < 6; ++nt) {
    v8f acc = {};
    const int colB = nt * 16 + lo;
    for (int s = 0; s < CHUNK / 4; ++s) {
      const int k0 = 4 * s + 2 * hi;
      v2f a, b;
      a.x = sA[lo * CHUNK + k0];
      a.y = sA[lo * CHUNK + k0 + 1];
      b.x = Wc[k0 * OUT + colB];
      b.y = Wc[(k0 + 1) * OUT + colB];
      acc = wmma4(a, b, acc);
    }
    const float bv = bias[c * OUT + nt * 16 + lo];
#pragma unroll
    for (int r = 0; r < 8; ++r)
      sO[(r + 8 * hi) * OUT + nt * 16 + lo] = acc[r] + bv;
  }
  __syncthreads();

  // LayerNorm over 96 features + LeakyReLU(0.01); lane pair (lo,hi) per row
  const int row = lo;
  float m = 0.0f;
  for (int j = 0; j < OUT; ++j) m += sO[row * OUT + j];
  m *= (1.0f / OUT);
  float v = 0.0f;
  for (int j = 0; j < OUT; ++j) {
    float d = sO[row * OUT + j] - m;
    v += d * d;
  }
  v *= (1.0f / OUT);
  const float rs = rsqrtf(v + EPS);
  for (int j = hi * 48; j < hi * 48 + 48; ++j) {
    float val = (sO[row * OUT + j] - m) * rs * g[c * OUT + j] + beta[c * OUT + j];
    val = val < 0.0f ? 0.01f * val : val;
    hh[(nb + row) * HID + c * OUT + j] = val;
  }
}

// ---------------- GcnSAGE layer: concat(hh, ah*norm) @ W[384,192] + b, LN, ReLU ----------------
// grid NN/16, block 32

__global__ __launch_bounds__(32) void gcn_kernel(
    const float* __restrict__ hh_in, const float* __restrict__ ah,
    const float* __restrict__ nrm, const float* __restrict__ W,
    const float* __restrict__ bias, const float* __restrict__ g,
    const float* __restrict__ beta, float* __restrict__ hh_out) {
  __shared__ alignas(16) float sA[16 * 2 * HID];  // 16 x 384 = 24 KB
  __shared__ float sO[16 * HID];                  // 16 x 192 = 12 KB
  const int lane = threadIdx.x;
  const int lo = lane & 15;
  const int hi = lane >> 4;
  const int nb = blockIdx.x * 16;

  // stage concat tile with b128 loads: per row 48 + 48 float4
  for (int idx = lane; idx < 16 * 96; idx += 32) {
    int r = idx / 96, q = idx % 96;
    float4 val;
    if (q < 48) {
      val = *reinterpret_cast<const float4*>(&hh_in[(nb + r) * HID + q * 4]);
    } else {
      val = *reinterpret_cast<const float4*>(&ah[(nb + r) * HID + (q - 48) * 4]);
      const float nr = nrm[nb + r];
      val.x *= nr; val.y *= nr; val.z *= nr; val.w *= nr;
    }
    *reinterpret_cast<float4*>(&sA[r * (2 * HID) + q * 4]) = val;
  }
  __syncthreads();

  for (int nt = 0; nt < 12; ++nt) {
    v8f acc = {};
    const int colB = nt * 16 + lo;
    for (int s = 0; s < (2 * HID) / 4; ++s) {
      const int k0 = 4 * s + 2 * hi;
      v2f a, b;
      a.x = sA[lo * (2 * HID) + k0];
      a.y = sA[lo * (2 * HID) + k0 + 1];
      b.x = W[k0 * HID + colB];
      b.y = W[(k0 + 1) * HID + colB];
      acc = wmma4(a, b, acc);
    }
    const float bv = bias[nt * 16 + lo];
#pragma unroll
    for (int r = 0; r < 8; ++r)
      sO[(r + 8 * hi) * HID + nt * 16 + lo] = acc[r] + bv;
  }
  __syncthreads();

  const int row = lo;
  float m = 0.0f;
  for (int j = 0; j < HID; ++j) m += sO[row * HID + j];
  m *= (1.0f / HID);
  float v = 0.0f;
  for (int j = 0; j < HID; ++j) {
    float d = sO[row * HID + j] - m;
    v += d * d;
  }
  v *= (1.0f / HID);
  const float rs = rsqrtf(v + EPS);
  for (int j = hi * 96; j < hi * 96 + 96; ++j) {
    float val = (sO[row * HID + j] - m) * rs * g[j] + beta[j];
    hh_out[(nb + row) * HID + j] = val > 0.0f ? val : 0.0f;
  }
}

// ---------------- edge MLP: relu((hh[src]*hh[dst]) @ W1 + b1) @ W2 + b2 ----------------
// grid EE/16, block 32 (one wave per 16-edge tile)

__global__ __launch_bounds__(32) void edge_kernel(
    const float* __restrict__ hh, const int* __restrict__ src,
    const int* __restrict__ dst, const float* __restrict__ W1,
    const float* __restrict__ b1, const float* __restrict__ W2,
    const float* __restrict__ b2, float* __restrict__ out) {
  __shared__ alignas(16) float sHe[16 * HID];  // 12 KB
  __shared__ float sZ[16 * OUT];               // 6 KB
  const int lane = threadIdx.x;
  const int lo = lane & 15;
  const int hi = lane >> 4;
  const int eb = blockIdx.x * 16;

  // stage he = hh[src]*hh[dst] with b128 loads: per edge 48 float4
  for (int idx = lane; idx < 16 * 48; idx += 32) {
    int e = idx / 48, q = idx % 48;
    const int s = src[eb + e], d = dst[eb + e];
    float4 a = *reinterpret_cast<const float4*>(&hh[s * HID + q * 4]);
    const float4 bq = *reinterpret_cast<const float4*>(&hh[d * HID + q * 4]);
    a.x *= bq.x; a.y *= bq.y; a.z *= bq.z; a.w *= bq.w;
    *reinterpret_cast<float4*>(&sHe[e * HID + q * 4]) = a;
  }
  __syncthreads();

  for (int nt = 0; nt < 6; ++nt) {
    v8f acc = {};
    const int colB = nt * 16 + lo;
    for (int s = 0; s < HID / 4; ++s) {
      const int k0 = 4 * s + 2 * hi;
      v2f a, b;
      a.x = sHe[lo * HID + k0];
      a.y = sHe[lo * HID + k0 + 1];
      b.x = W1[k0 * OUT + colB];
      b.y = W1[(k0 + 1) * OUT + colB];
      acc = wmma4(a, b, acc);
    }
    const float bv = b1[nt * 16 + lo];
#pragma unroll
    for (int r = 0; r < 8; ++r) {
      float z = acc[r] + bv;
      sZ[(r + 8 * hi) * OUT + nt * 16 + lo] = z > 0.0f ? z : 0.0f;
    }
  }
  __syncthreads();

  // final 96 -> 2: one (edge, class) pair per lane
  const int e = lane >> 1;
  const int cls = lane & 1;
  float sum = b2[cls];
  for (int j = 0; j < OUT; ++j) sum += sZ[e * OUT + j] * W2[j * 2 + cls];
  out[(eb + e) * 2 + cls] = sum;
}

// ---------------- driver ----------------

extern "C" void kernel_launch(void* const* d_in, const int* in_sizes, int n_in,
                              void* d_out, int out_size, void* d_ws, size_t ws_size,
                              hipStream_t stream) {
  const float* h        = (const float*)d_in[0];
  const float* dist     = (const float*)d_in[1];
  const int*   src      = (const int*)d_in[2];
  const int*   dst      = (const int*)d_in[3];
  const float* proj_W   = (const float*)d_in[4];
  const float* proj_b   = (const float*)d_in[5];
  const float* proj_g   = (const float*)d_in[6];
  const float* proj_beta= (const float*)d_in[7];
  const float* gcn_W    = (const float*)d_in[8];
  const float* gcn_b    = (const float*)d_in[9];
  const float* gcn_g    = (const float*)d_in[10];
  const float* gcn_beta = (const float*)d_in[11];
  const float* ep_W1    = (const float*)d_in[12];
  const float* ep_b1    = (const float*)d_in[13];
  const float* ep_W2    = (const float*)d_in[14];
  const float* ep_b2    = (const float*)d_in[15];
  float* out = (float*)d_out;

  // workspace layout
  float* hhA      = (float*)d_ws;                 // NN*HID f32
  float* hhB      = hhA + (size_t)NN * HID;       // NN*HID f32
  float* ah       = hhB + (size_t)NN * HID;       // NN*HID f32
  float* nrm      = ah + (size_t)NN * HID;        // NN f32
  float* csr_w    = nrm + NN;                     // EE f32
  int*   degi     = (int*)(csr_w + EE);           // NN i32
  int*   rowStart = degi + NN;                    // NN+1 i32
  int*   cursor   = rowStart + (NN + 1);          // NN i32
  int*   csr_src  = cursor + NN;                  // EE i32

  // ---- CSR build + degree norm ----
  zero_i_kernel<<<(NN + 255) / 256, 256, 0, stream>>>(degi, NN);
  zero_i_kernel<<<(NN + 255) / 256, 256, 0, stream>>>(cursor, NN);
  degi_kernel<<<(EE + 255) / 256, 256, 0, stream>>>(dst, degi);
  norm_kernel<<<(NN + 255) / 256, 256, 0, stream>>>(degi, nrm);
  scan_kernel<<<1, 1024, 0, stream>>>(degi, rowStart);
  fill_kernel<<<(EE + 255) / 256, 256, 0, stream>>>(src, dst, dist, rowStart,
                                                    cursor, csr_src, csr_w);

  // ---- input projector ----
  proj_kernel<<<dim3(NN / 16, 2), 32, 0, stream>>>(h, proj_W, proj_b, proj_g,
                                                   proj_beta, hhA);

  // ---- 2 GcnSAGE layers (ping-pong hhA <-> hhB) ----
  const float* cur = hhA;
  float* nxt = hhB;
  for (int l = 0; l < 2; ++l) {
    gather_kernel<<<NN, HID, 0, stream>>>(cur, rowStart, csr_src, csr_w, ah);
    gcn_kernel<<<NN / 16, 32, 0, stream>>>(cur, ah, nrm,
                                           gcn_W + (size_t)l * 2 * HID * HID,
                                           gcn_b + (size_t)l * HID,
                                           gcn_g + (size_t)l * HID,
                                           gcn_beta + (size_t)l * HID, nxt);
    float* t = (float*)cur;
    cur = nxt;
    nxt = t;
  }

  // ---- edge scoring ----
  edge_kernel<<<EE / 16, 32, 0, stream>>>(cur, src, dst, ep_W1, ep_b1, ep_W2,
                                          ep_b2, out);
}